// ScaledDotProductAttention_4647154614503
// MI455X (gfx1250) — compile-verified
//
#include <hip/hip_runtime.h>
#include <math.h>

// ---------------------------------------------------------------------------
// Flash-style scaled-dot-product attention for MI455X (gfx1250, wave32).
//   keys  : [B=32, S=2048, H=1024] f32   (values == keys)
//   query : [B=32, H=1024] f32
//   out   : attn_values [B,H] f32  ++  attn [B,S] f32   (concatenated)
//
// Bandwidth-bound (268 MB of keys vs 0.27 GFLOP) -> single streaming pass over
// keys with online softmax. Scores computed with V_WMMA_F32_16X16X4_F32:
//   A (16x4 f32): lane l<16 -> row l, k={k0,k0+1}; lane 16+l -> row l, k={k0+2,k0+3}
//   B (4x16 f32): query chunk replicated across all 16 columns (same half-wave
//                 k-split as A), loaded from LDS (broadcast reads).
//   C/D (16x16 f32): VGPR j, lanes 0-15 -> M=j ; lanes 16-31 -> M=8+j.
// All 16 columns of D are identical == the 16 row scores.
// While phase B (cache-hit weighted accumulation) runs, the next tile is
// pulled toward WGP$/L2 with global_prefetch_b8.
// ---------------------------------------------------------------------------

typedef float v2f __attribute__((ext_vector_type(2)));
typedef float v8f __attribute__((ext_vector_type(8)));

#define B_      32
#define S_      2048
#define H_      1024
#define NSPLIT  16
#define WAVES   4
#define ROWS_PER_WG   (S_ / NSPLIT)         // 128
#define ROWS_PER_WAVE (ROWS_PER_WG / WAVES) // 32
#define TILES_PER_WAVE (ROWS_PER_WAVE / 16) // 2
#define SCALE   0.03125f                    // 1/sqrt(1024)

// workspace layout (floats)
#define OFF_SC   0                            // B*S scaled raw scores
#define OFF_M    (B_ * S_)                    // B*NSPLIT partial max
#define OFF_L    (OFF_M + B_ * NSPLIT)        // B*NSPLIT partial sum
#define OFF_ACC  (OFF_L + B_ * NSPLIT)        // B*NSPLIT*H partial weighted sums
#define OFF_GM   (OFF_ACC + B_ * NSPLIT * H_) // B global max
#define OFF_GL   (OFF_GM + B_)                // B global denom

__global__ __launch_bounds__(WAVES * 32)
void attn_partial_kernel(const float* __restrict__ keys,
                         const float* __restrict__ query,
                         float* __restrict__ ws) {
    const int wg    = blockIdx.x;           // 0 .. B*NSPLIT-1
    const int b     = wg / NSPLIT;
    const int split = wg % NSPLIT;
    const int tid   = threadIdx.x;
    const int wave  = tid >> 5;
    const int lane  = tid & 31;
    const int half  = lane >> 4;            // 0: k/k+1  1: k+2/k+3
    const int mrow  = lane & 15;

    float* sc_ws  = ws + OFF_SC;
    float* m_ws   = ws + OFF_M;
    float* l_ws   = ws + OFF_L;
    float* acc_ws = ws + OFF_ACC;

    __shared__ float q_lds[H_];
    __shared__ float acc_lds[WAVES * H_];
    __shared__ float ml_lds[2 * WAVES];

    // stage query[b] into LDS: 128 threads x 2 float4
    {
        const float4* qg = (const float4*)(query + (size_t)b * H_);
        float4* ql = (float4*)q_lds;
        ql[tid]       = qg[tid];
        ql[tid + 128] = qg[tid + 128];
    }
    __syncthreads();

    const float* kb = keys + (size_t)b * S_ * H_;

    float  m = -INFINITY;
    float  l = 0.0f;
    float4 acc4[8];
#pragma unroll
    for (int i = 0; i < 8; ++i) acc4[i] = make_float4(0.f, 0.f, 0.f, 0.f);

    for (int t = 0; t < TILES_PER_WAVE; ++t) {
        const int r0 = split * ROWS_PER_WG + wave * ROWS_PER_WAVE + t * 16;

        // ---- Phase A: 16 scores via fp32 WMMA, K accumulated over H ----
        v8f c = {};
        const float* arow = kb + (size_t)(r0 + mrow) * H_ + (half << 1);
        const float* qoff = q_lds + (half << 1);
#pragma unroll 8
        for (int k0 = 0; k0 < H_; k0 += 4) {
            v2f a  = *(const v2f*)(arow + k0);
            v2f bq = *(const v2f*)(qoff + k0);
            c = __builtin_amdgcn_wmma_f32_16x16x4_f32(
                    false, a, false, bq, (short)0, c, false, false);
        }

        // extract 16 wave-uniform scaled scores
        float s[16];
#pragma unroll
        for (int j = 0; j < 8; ++j) {
            s[j]     = __shfl(c[j], 0)  * SCALE;   // rows 0..7
            s[j + 8] = __shfl(c[j], 16) * SCALE;   // rows 8..15
        }
        if (lane == 0) {
#pragma unroll
            for (int j = 0; j < 16; ++j)
                sc_ws[(size_t)b * S_ + r0 + j] = s[j];
        }

        // prefetch next tile (64KB = 512 x 128B lines, 32 lanes x 16 insts)
        // while phase B below runs out of cache.
        if (t + 1 < TILES_PER_WAVE) {
            const float* nxt = kb + (size_t)(r0 + 16) * H_;
#pragma unroll
            for (int i = 0; i < 16; ++i)
                __builtin_prefetch(nxt + (size_t)(i * 32 + lane) * 32, 0, 3);
        }

        // ---- online softmax update ----
        float tmax = s[0];
#pragma unroll
        for (int j = 1; j < 16; ++j) tmax = fmaxf(tmax, s[j]);
        const float newm = fmaxf(m, tmax);
        const float corr = __expf(m - newm);       // exp(-inf)=0 first tile
        float e[16];
        float esum = 0.f;
#pragma unroll
        for (int j = 0; j < 16; ++j) { e[j] = __expf(s[j] - newm); esum += e[j]; }
        l = l * corr + esum;
#pragma unroll
        for (int i = 0; i < 8; ++i) {
            acc4[i].x *= corr; acc4[i].y *= corr;
            acc4[i].z *= corr; acc4[i].w *= corr;
        }
        m = newm;

        // ---- Phase B: acc += e[j] * K[row] (coalesced; tile hot in WGP$/L2)
#pragma unroll 4
        for (int j = 0; j < 16; ++j) {
            const float4* rp = (const float4*)(kb + (size_t)(r0 + j) * H_);
            const float w = e[j];
#pragma unroll
            for (int i = 0; i < 8; ++i) {
                float4 kv = rp[i * 32 + lane];
                acc4[i].x += w * kv.x; acc4[i].y += w * kv.y;
                acc4[i].z += w * kv.z; acc4[i].w += w * kv.w;
            }
        }
    }

    // ---- combine the 4 wave partials inside the workgroup ----
#pragma unroll
    for (int i = 0; i < 8; ++i)
        *(float4*)&acc_lds[wave * H_ + i * 128 + lane * 4] = acc4[i];
    if (lane == 0) { ml_lds[wave] = m; ml_lds[WAVES + wave] = l; }
    __syncthreads();

    const float m0 = ml_lds[0], m1 = ml_lds[1], m2 = ml_lds[2], m3 = ml_lds[3];
    const float M  = fmaxf(fmaxf(m0, m1), fmaxf(m2, m3));
    const float w0 = __expf(m0 - M), w1 = __expf(m1 - M);
    const float w2 = __expf(m2 - M), w3 = __expf(m3 - M);
    const float L  = w0 * ml_lds[4] + w1 * ml_lds[5] +
                     w2 * ml_lds[6] + w3 * ml_lds[7];

    float* accw = acc_ws + ((size_t)b * NSPLIT + split) * H_;
    for (int d = tid; d < H_; d += WAVES * 32) {
        float v = w0 * acc_lds[d] + w1 * acc_lds[H_ + d] +
                  w2 * acc_lds[2 * H_ + d] + w3 * acc_lds[3 * H_ + d];
        accw[d] = v;
    }
    if (tid == 0) {
        m_ws[b * NSPLIT + split] = M;
        l_ws[b * NSPLIT + split] = L;
    }
}

__global__ __launch_bounds__(256)
void attn_combine_kernel(const float* __restrict__ ws, float* __restrict__ out) {
    const int b   = blockIdx.x;
    const int tid = threadIdx.x;
    const float* m_ws   = ws + OFF_M  + b * NSPLIT;
    const float* l_ws   = ws + OFF_L  + b * NSPLIT;
    const float* acc_ws = ws + OFF_ACC + (size_t)b * NSPLIT * H_;
    float* gm = (float*)(ws + OFF_GM);
    float* gl = (float*)(ws + OFF_GL);

    float M = -INFINITY;
#pragma unroll
    for (int i = 0; i < NSPLIT; ++i) M = fmaxf(M, m_ws[i]);
    float w[NSPLIT];
    float L = 0.f;
#pragma unroll
    for (int i = 0; i < NSPLIT; ++i) { w[i] = __expf(m_ws[i] - M); L += w[i] * l_ws[i]; }
    const float invL = 1.0f / L;

    for (int d = tid; d < H_; d += 256) {
        float v = 0.f;
#pragma unroll
        for (int i = 0; i < NSPLIT; ++i) v += w[i] * acc_ws[(size_t)i * H_ + d];
        out[(size_t)b * H_ + d] = v * invL;
    }
    if (tid == 0) { gm[b] = M; gl[b] = L; }
}

__global__ __launch_bounds__(256)
void attn_probs_kernel(const float* __restrict__ ws, float* __restrict__ out) {
    const int idx = blockIdx.x * 256 + threadIdx.x;   // 0 .. B*S-1
    const int b   = idx >> 11;                        // S = 2048
    const float sc = ws[OFF_SC + idx];
    const float M  = ws[OFF_GM + b];
    const float L  = ws[OFF_GL + b];
    out[idx] = __expf(sc - M) / L;
}

extern "C" void kernel_launch(void* const* d_in, const int* in_sizes, int n_in,
                              void* d_out, int out_size, void* d_ws, size_t ws_size,
                              hipStream_t stream) {
    const float* keys  = (const float*)d_in[0];   // [B,S,H]
    const float* query = (const float*)d_in[1];   // [B,H]
    (void)in_sizes; (void)n_in; (void)out_size; (void)ws_size;

    float* out = (float*)d_out;                   // [B*H] ++ [B*S]
    float* ws  = (float*)d_ws;                    // needs ~2.4 MB

    attn_partial_kernel<<<B_ * NSPLIT, WAVES * 32, 0, stream>>>(keys, query, ws);
    attn_combine_kernel<<<B_, 256, 0, stream>>>(ws, out);
    attn_probs_kernel<<<(B_ * S_) / 256, 256, 0, stream>>>(ws, out + B_ * H_);
}